// CompositeSegmentationLoss_58471684768278
// MI455X (gfx1250) — compile-verified
//
#include <hip/hip_runtime.h>
#include <hip/hip_bf16.h>

// ---------------------------------------------------------------------------
// CompositeSegmentationLoss for MI455X (gfx1250, wave32).
//
// Roofline: inputs are 6 MB (0.3 us at 23.3 TB/s HBM). The only real compute
// is the EDT row lower-envelope (min-plus, ~1 GFLOP total) which must stay on
// VALU. The dice/tversky/attention reductions are mapped onto
// V_WMMA_F32_16X16X4_F32 (B = ones => fp32-exact row sums, 64 elems/wave/op),
// keeping the matrix pipes busy and the sums bit-deterministic.
// ---------------------------------------------------------------------------

typedef __attribute__((ext_vector_type(2))) float v2f;
typedef __attribute__((ext_vector_type(8))) float v8f;

#define BATCH 8
#define HH 256
#define WW 256
#define NPIX (HH * WW)          // 65536
#define BIGD 1.0e4f
#define EPSL 1e-8f

// ws layout (in floats):
//   [0,            524288)  g2m  : squared column-distance, mask
//   [524288,      1048576)  g2c  : squared column-distance, ~mask
//   [1048576,     1048832)  partialsA : 64 blocks x 4 sums
//   [1048832,     1050880)  rowPartials : 2048 rows
#define WS_G2M 0
#define WS_G2C 524288
#define WS_PA  1048576
#define WS_ROW 1048832

// ---------------------------------------------------------------------------
// Kernel 1: WMMA-based per-batch reductions.
// grid = 64 blocks (8 per batch), 256 threads (8 waves). Each wave folds 1024
// elements into four resident 16x16 f32 accumulators via wmma with B = ones:
//   D[m][n] += sum_k A[m][k]  -> every column replicates the sum; divide by 16.
// ---------------------------------------------------------------------------
__global__ void reduce_wmma_kernel(const float* __restrict__ probs,
                                   const int*   __restrict__ targets,
                                   const float* __restrict__ attn,
                                   float*       __restrict__ partials) {
  const int b    = blockIdx.x >> 3;
  const int blk  = blockIdx.x & 7;
  const int wave = threadIdx.x >> 5;
  const int lane = threadIdx.x & 31;
  const int base = b * NPIX + blk * 8192 + wave * 1024;

  v8f accP = {}; v8f accT = {}; v8f accPT = {}; v8f accD = {};
  const v2f ones = {1.0f, 1.0f};

  for (int it = 0; it < 16; ++it) {
    const int idx = base + it * 64 + lane * 2;
    v2f  p  = *(const v2f*)(probs + idx);
    int2 ti = *(const int2*)(targets + idx);
    v2f  a  = *(const v2f*)(attn + idx);
    v2f  t  = { (float)ti.x, (float)ti.y };
    v2f  pt = { p.x * t.x, p.y * t.y };
    v2f  d0 = { a.x - t.x, a.y - t.y };
    v2f  dd = { d0.x * d0.x, d0.y * d0.y };
    // (neg_a, A, neg_b, B, c_mod, C, reuse_a, reuse_b)
    accP  = __builtin_amdgcn_wmma_f32_16x16x4_f32(false, p,  false, ones, (short)0, accP,  false, false);
    accT  = __builtin_amdgcn_wmma_f32_16x16x4_f32(false, t,  false, ones, (short)0, accT,  false, false);
    accPT = __builtin_amdgcn_wmma_f32_16x16x4_f32(false, pt, false, ones, (short)0, accPT, false, false);
    accD  = __builtin_amdgcn_wmma_f32_16x16x4_f32(false, dd, false, ones, (short)0, accD,  false, false);
  }

  float sP = 0.f, sT = 0.f, sPT = 0.f, sD = 0.f;
  for (int r = 0; r < 8; ++r) { sP += accP[r]; sT += accT[r]; sPT += accPT[r]; sD += accD[r]; }

  __shared__ float red[4][256];
  red[0][threadIdx.x] = sP;  red[1][threadIdx.x] = sT;
  red[2][threadIdx.x] = sPT; red[3][threadIdx.x] = sD;
  __syncthreads();
  if (threadIdx.x < 4) {
    float s = 0.f;
    for (int i = 0; i < 256; ++i) s += red[threadIdx.x][i];   // fixed order: deterministic
    partials[blockIdx.x * 4 + threadIdx.x] = s * (1.0f / 16.0f);
  }
}

// ---------------------------------------------------------------------------
// Kernel 2: column pass of the exact EDT (matches the reference's max/min
// scans with the BIG sentinel), for mask and complement simultaneously.
// One thread per (batch, column): forward scan stores y-up, backward scan
// combines with dn-y, clamps at BIG and squares. grid = 8 x 256 threads.
// ---------------------------------------------------------------------------
__global__ void colpass_kernel(const int* __restrict__ targets,
                               float* __restrict__ g2m,
                               float* __restrict__ g2c) {
  const int gid = blockIdx.x * blockDim.x + threadIdx.x;  // 0..2047
  const int b = gid >> 8;
  const int x = gid & 255;
  const int* tb = targets + b * NPIX + x;
  float* gm = g2m + b * NPIX + x;
  float* gc = g2c + b * NPIX + x;

  float upm = -BIGD, upc = -BIGD;
  for (int y = 0; y < HH; ++y) {
    const bool m = (tb[y * WW] != 0);
    const float fy = (float)y;
    if (m) upm = fy; else upc = fy;           // inclusive max-scan of (mask? y : -BIG)
    gm[y * WW] = fminf(fy - upm, BIGD);
    gc[y * WW] = fminf(fy - upc, BIGD);
  }
  float dnm = 2.0f * BIGD, dnc = 2.0f * BIGD;
  for (int y = HH - 1; y >= 0; --y) {
    const bool m = (tb[y * WW] != 0);
    const float fy = (float)y;
    if (m) dnm = fy; else dnc = fy;           // inclusive reverse min-scan
    float g1 = fminf(fminf(gm[y * WW], dnm - fy), BIGD);
    float g2 = fminf(fminf(gc[y * WW], dnc - fy), BIGD);
    gm[y * WW] = g1 * g1;
    gc[y * WW] = g2 * g2;
  }
}

// ---------------------------------------------------------------------------
// Kernel 3: row lower-envelope (brute force, exactly the reference's
// min_{x'} g^2[x'] + (x-x')^2), sqrt(d2m)+sqrt(d2c), weight by probs and
// block-reduce. One block per (batch,row); g^2 rows staged in LDS so the
// 256-iteration inner loop is broadcast reads + v_fma/v_min.
// ---------------------------------------------------------------------------
__global__ void rowpass_kernel(const float* __restrict__ probs,
                               const float* __restrict__ g2m,
                               const float* __restrict__ g2c,
                               float* __restrict__ rowPartials) {
  const int row = blockIdx.x;          // b*256 + y
  const int x   = threadIdx.x;
  __shared__ float lm[WW];
  __shared__ float lc[WW];
  __shared__ float red[WW];

  __builtin_prefetch(probs + row * WW + x, 0, 3);   // global_prefetch_b8
  lm[x] = g2m[row * WW + x];
  lc[x] = g2c[row * WW + x];
  __syncthreads();

  float dm = 3.4e38f, dc = 3.4e38f;
  #pragma unroll 8
  for (int xp = 0; xp < WW; ++xp) {
    const float dx  = (float)(x - xp);
    const float dx2 = dx * dx;                      // exact (integers < 2^24)
    dm = fminf(dm, lm[xp] + dx2);
    dc = fminf(dc, lc[xp] + dx2);
  }
  const float sdt = sqrtf(dm) + sqrtf(dc);          // |SDT| = pos + neg
  red[x] = probs[row * WW + x] * sdt;
  __syncthreads();
  for (int s = 128; s > 0; s >>= 1) {               // fixed-tree: deterministic
    if (x < s) red[x] += red[x + s];
    __syncthreads();
  }
  if (x == 0) rowPartials[row] = red[0];
}

// ---------------------------------------------------------------------------
// Kernel 4: single-block finalize. Fixed-order combines, dice/tversky
// formulas, any_fg gate (sum(t) > 0.5), loss weights.
// ---------------------------------------------------------------------------
__global__ void finalize_kernel(const float* __restrict__ partialsA,
                                const float* __restrict__ rowPartials,
                                float* __restrict__ out) {
  __shared__ float S[BATCH][4];
  __shared__ float Rb[BATCH];
  const int tid = threadIdx.x;
  if (tid < 32) {
    const int b = tid >> 2, q = tid & 3;
    float s = 0.f;
    for (int blk = 0; blk < 8; ++blk) s += partialsA[(b * 8 + blk) * 4 + q];
    S[b][q] = s;
  }
  if (tid >= 32 && tid < 32 + BATCH) {
    const int b = tid - 32;
    float s = 0.f;
    for (int y = 0; y < HH; ++y) s += rowPartials[b * HH + y];
    Rb[b] = s;
  }
  __syncthreads();
  if (tid == 0) {
    float dl = 0.f, tl = 0.f, bl = 0.f, al = 0.f;
    for (int b = 0; b < BATCH; ++b) {
      const float sumP = S[b][0], sumT = S[b][1], inter = S[b][2], diff2 = S[b][3];
      dl += (2.f * inter + EPSL) / (sumP + sumT + EPSL);
      const float fn = sumT - inter;     // sum t*(1-p)
      const float fp = sumP - inter;     // sum (1-t)*p
      tl += (inter + EPSL) / (inter + 0.3f * fn + 0.7f * fp + EPSL);
      bl += (sumT > 0.5f) ? Rb[b] : 0.f; // any_fg gate (sum t is exact)
      al += diff2;
    }
    dl = 1.f - dl / (float)BATCH;
    tl = 1.f - tl / (float)BATCH;
    const float invN = 1.f / (float)(BATCH * NPIX);
    out[0] = 1.0f * dl + 0.7f * tl + 0.5f * (bl * invN) + 0.3f * (al * invN);
  }
}

// ---------------------------------------------------------------------------
extern "C" void kernel_launch(void* const* d_in, const int* in_sizes, int n_in,
                              void* d_out, int out_size, void* d_ws, size_t ws_size,
                              hipStream_t stream) {
  const float* probs   = (const float*)d_in[0];
  const int*   targets = (const int*)d_in[1];
  const float* attn    = (const float*)d_in[2];
  float* out = (float*)d_out;
  float* ws  = (float*)d_ws;

  float* g2m = ws + WS_G2M;
  float* g2c = ws + WS_G2C;
  float* pa  = ws + WS_PA;
  float* rp  = ws + WS_ROW;

  reduce_wmma_kernel<<<64, 256, 0, stream>>>(probs, targets, attn, pa);
  colpass_kernel<<<8, 256, 0, stream>>>(targets, g2m, g2c);
  rowpass_kernel<<<BATCH * HH, 256, 0, stream>>>(probs, g2m, g2c, rp);
  finalize_kernel<<<1, 64, 0, stream>>>(pa, rp, out);
}